// LightGCN_20014547599450
// MI455X (gfx1250) — compile-verified
//
#include <hip/hip_runtime.h>

// ---- problem constants (from the reference) ----
#define N_USERS   200000
#define N_ITEMS   50000
#define N_NODES   (N_USERS + N_ITEMS)   // 250000
#define DIM       64
#define NNZ       4000000               // 2 * N_INTER
#define BATCH     4096
#define NELEM     (N_NODES * DIM)       // 16,000,000 floats = 64 MB

typedef __attribute__((ext_vector_type(2))) float v2f;
typedef __attribute__((ext_vector_type(8))) float v8f;

// ---------------- elementwise helpers (float4-vectorized) ----------------

__global__ void lgcn_copy_f4(const float4* __restrict__ src, float4* __restrict__ dst, int n4) {
    int i = blockIdx.x * blockDim.x + threadIdx.x;
    if (i < n4) dst[i] = src[i];
}

__global__ void lgcn_zero_f4(float4* __restrict__ dst, int n4) {
    int i = blockIdx.x * blockDim.x + threadIdx.x;
    if (i < n4) dst[i] = make_float4(0.f, 0.f, 0.f, 0.f);
}

__global__ void lgcn_add_f4(float4* __restrict__ acc, const float4* __restrict__ e, int n4) {
    int i = blockIdx.x * blockDim.x + threadIdx.x;
    if (i < n4) {
        float4 a = acc[i];
        float4 b = e[i];
        a.x += b.x; a.y += b.y; a.z += b.z; a.w += b.w;
        acc[i] = a;
    }
}

// ---------------- SpMM: y[row] += val * x[col], 16 lanes per edge ----------------
// Each lane handles a float4 (4 dims) of one edge: vectorized 16B gather from the
// (L2-resident, 64MB) source table, then 4 hardware fp32 atomic adds
// (global_atomic_add_f32 via unsafeAtomicAdd — no CAS-loop expansion) into the
// (L2-resident) destination table.

__global__ void lgcn_spmm_scatter(const float* __restrict__ x,
                                  const int*   __restrict__ rows,
                                  const int*   __restrict__ cols,
                                  const float* __restrict__ vals,
                                  float*       __restrict__ y) {
    unsigned int tid  = blockIdx.x * blockDim.x + threadIdx.x;
    unsigned int edge = tid >> 4;
    unsigned int lane = tid & 15u;
    if (edge >= NNZ) return;

    int   c = cols[edge];
    int   r = rows[edge];
    float v = vals[edge];

    const float4* src = (const float4*)(x + c * DIM) + lane;
    float4 s = *src;

    float* dst = y + r * DIM + lane * 4;
    unsafeAtomicAdd(dst + 0, v * s.x);
    unsafeAtomicAdd(dst + 1, v * s.y);
    unsafeAtomicAdd(dst + 2, v * s.z);
    unsafeAtomicAdd(dst + 3, v * s.w);
}

// ---------------- final gather + /4 via WMMA f32 16x16x4 ----------------
// One wave32 per 16-row tile. Computes Y(16x64) = X(16x64) * (0.25 * I64) with
// 4 col-tiles x 4 chained V_WMMA_F32_16X16X4_F32 each.
// A (16x4 f32, 2 VGPR/lane): lanes 0-15 hold rows M=0..15 K={0,1}; lanes 16-31 K={2,3}.
// B (4x16 f32, 2 VGPR/lane): vgpr v, lane L -> row k = v + 2*(L/16), col = L%16.
// C/D (16x16 f32, 8 VGPR/lane): vgpr r, lane L -> M = r + 8*(L/16), N = L%16.

__global__ void lgcn_gather_final_wmma(const float* __restrict__ acc,
                                       const long long* __restrict__ users,
                                       const long long* __restrict__ pos_items,
                                       const long long* __restrict__ neg_items,
                                       float* __restrict__ out) {
    unsigned int gtid = blockIdx.x * blockDim.x + threadIdx.x;
    unsigned int wave = gtid >> 5;          // 768 waves total: 3 outputs * 256 tiles
    unsigned int L    = gtid & 31u;
    unsigned int o    = wave / (BATCH / 16);        // which output array [0,3)
    unsigned int tile = wave % (BATCH / 16);        // 16-row tile within the batch
    unsigned int b0   = tile * 16;
    unsigned int half = L >> 4;                     // 0: lanes 0-15, 1: lanes 16-31
    unsigned int mrow = L & 15u;                    // row-in-tile this lane serves
    unsigned int b    = b0 + mrow;

    long long rowidx;
    if (o == 0)      rowidx = users[b];
    else if (o == 1) rowidx = (long long)N_USERS + pos_items[b];
    else             rowidx = (long long)N_USERS + neg_items[b];

    const float* xrow = acc + rowidx * DIM;

    #pragma unroll
    for (int t = 0; t < 4; ++t) {           // 16-column tile of the 64-dim output
        v8f c = {};
        #pragma unroll
        for (int m4 = 0; m4 < 4; ++m4) {    // K-chunk (4 of them per col-tile)
            int m = 4 * t + m4;             // global K-chunk: covers dims 4m..4m+3
            // A: this lane's 2 K-values of row `mrow`
            v2f a = *(const v2f*)(xrow + 4 * m + 2 * half);
            // B: 0.25 * selection — B[k][col] = 0.25 iff col == 4*m4 + k
            int k0 = 2 * (int)half;
            v2f bm;
            bm.x = ((int)mrow == 4 * m4 + k0 + 0) ? 0.25f : 0.0f;
            bm.y = ((int)mrow == 4 * m4 + k0 + 1) ? 0.25f : 0.0f;
            c = __builtin_amdgcn_wmma_f32_16x16x4_f32(
                    /*neg_a=*/false, a, /*neg_b=*/false, bm,
                    /*c_mod=*/(short)0, c, /*reuse_a=*/false, /*reuse_b=*/false);
        }
        // store D: lane L, vgpr r -> row (b0 + r + 8*half), col (16*t + mrow)
        #pragma unroll
        for (int r = 0; r < 8; ++r) {
            unsigned int orow = b0 + r + 8 * half;
            out[((size_t)o * BATCH + orow) * DIM + 16 * t + mrow] = c[r];
        }
    }
}

// ---------------- init-embedding gather (outputs 3..5) ----------------

__global__ void lgcn_gather_init(const float* __restrict__ emb,
                                 const long long* __restrict__ users,
                                 const long long* __restrict__ pos_items,
                                 const long long* __restrict__ neg_items,
                                 float* __restrict__ out) {
    unsigned int tid  = blockIdx.x * blockDim.x + threadIdx.x;  // 3*4096*16 threads
    unsigned int o    = tid / (BATCH * 16);
    unsigned int rem  = tid % (BATCH * 16);
    unsigned int b    = rem >> 4;
    unsigned int lane = rem & 15u;

    long long idx;
    if (o == 0)      idx = users[b];
    else if (o == 1) idx = (long long)N_USERS + pos_items[b];
    else             idx = (long long)N_USERS + neg_items[b];

    const float4* s = (const float4*)(emb + idx * DIM) + lane;
    float4*       d = (float4*)(out + ((size_t)(3 + o) * BATCH + b) * DIM) + lane;
    *d = *s;
}

// ---------------- launch ----------------

extern "C" void kernel_launch(void* const* d_in, const int* in_sizes, int n_in,
                              void* d_out, int out_size, void* d_ws, size_t ws_size,
                              hipStream_t stream) {
    const float*     embed = (const float*)d_in[0];
    const int*       rows  = (const int*)d_in[1];
    const int*       cols  = (const int*)d_in[2];
    const float*     vals  = (const float*)d_in[3];
    const long long* users = (const long long*)d_in[4];
    const long long* pos   = (const long long*)d_in[5];
    const long long* neg   = (const long long*)d_in[6];
    float* out = (float*)d_out;

    float* acc = (float*)d_ws;          // 64 MB
    float* eP  = acc + NELEM;           // 64 MB
    float* eQ  = eP + NELEM;            // 64 MB

    const int n4       = NELEM / 4;                 // 4,000,000 float4s
    const int eltBlk   = 256;
    const int eltGrid  = (n4 + eltBlk - 1) / eltBlk;
    const int spmmBlk  = 256;
    const int spmmGrid = (NNZ * 16) / spmmBlk;      // 64M threads

    // acc = E0
    lgcn_copy_f4<<<eltGrid, eltBlk, 0, stream>>>((const float4*)embed, (float4*)acc, n4);

    // layer 1: eP = A*E0 ; acc += eP
    lgcn_zero_f4<<<eltGrid, eltBlk, 0, stream>>>((float4*)eP, n4);
    lgcn_spmm_scatter<<<spmmGrid, spmmBlk, 0, stream>>>(embed, rows, cols, vals, eP);
    lgcn_add_f4<<<eltGrid, eltBlk, 0, stream>>>((float4*)acc, (const float4*)eP, n4);

    // layer 2: eQ = A*eP ; acc += eQ
    lgcn_zero_f4<<<eltGrid, eltBlk, 0, stream>>>((float4*)eQ, n4);
    lgcn_spmm_scatter<<<spmmGrid, spmmBlk, 0, stream>>>(eP, rows, cols, vals, eQ);
    lgcn_add_f4<<<eltGrid, eltBlk, 0, stream>>>((float4*)acc, (const float4*)eQ, n4);

    // layer 3: eP = A*eQ ; acc += eP
    lgcn_zero_f4<<<eltGrid, eltBlk, 0, stream>>>((float4*)eP, n4);
    lgcn_spmm_scatter<<<spmmGrid, spmmBlk, 0, stream>>>(eQ, rows, cols, vals, eP);
    lgcn_add_f4<<<eltGrid, eltBlk, 0, stream>>>((float4*)acc, (const float4*)eP, n4);

    // final gathers: mean = acc/4 via WMMA (outputs 0..2), init embeddings (3..5)
    lgcn_gather_final_wmma<<<96, 256, 0, stream>>>(acc, users, pos, neg, out);
    lgcn_gather_init<<<(3 * BATCH * 16) / 256, 256, 0, stream>>>(embed, users, pos, neg, out);
}